// LightGCN_71064528880082
// MI455X (gfx1250) — compile-verified
//
#include <hip/hip_runtime.h>

typedef float v2f __attribute__((ext_vector_type(2)));
typedef float v8f __attribute__((ext_vector_type(8)));

#define UDIM 8192
#define IDIM 8192
#define DDIM 64
#define NROWS (UDIM + IDIM)
#define NELEM (NROWS * DDIM) // 1048576 floats per embedding buffer

// ---------------------------------------------------------------------------
// init: cur = acc = concat(users, items); next = 0; loss = 0
// ---------------------------------------------------------------------------
__global__ __launch_bounds__(256) void lg_init(const float* __restrict__ users,
                                               const float* __restrict__ items,
                                               float* __restrict__ cur,
                                               float* __restrict__ acc,
                                               float* __restrict__ nxt,
                                               float* __restrict__ loss) {
  int i = blockIdx.x * 256 + threadIdx.x;
  float v = (i < UDIM * DDIM) ? users[i] : items[i - UDIM * DDIM];
  cur[i] = v;
  acc[i] = v;
  nxt[i] = 0.0f;
  if (i == 0) loss[0] = 0.0f;
}

// ---------------------------------------------------------------------------
// SpMM (COO): one wave32 per edge; lane handles float2 of the 64-wide row.
// y[row] += val * x[col]  via global_atomic_add_f32 (y is L2-resident, 4 MB).
// ---------------------------------------------------------------------------
__global__ __launch_bounds__(256) void lg_spmm(const int* __restrict__ erow,
                                               const int* __restrict__ ecol,
                                               const float* __restrict__ eval,
                                               const float* __restrict__ x,
                                               float* __restrict__ y, int E) {
  int gid  = blockIdx.x * 256 + threadIdx.x;
  int e    = gid >> 5;
  int lane = gid & 31;
  if (e >= E) return;
  int   r = erow[e];
  int   c = ecol[e];
  float v = eval[e];
  v2f xv = *(const v2f*)(x + (size_t)c * DDIM + lane * 2);
  float* yd = y + (size_t)r * DDIM + lane * 2;
  atomicAdd(yd,     v * xv.x);
  atomicAdd(yd + 1, v * xv.y);
}

// ---------------------------------------------------------------------------
// acc += next; zero the buffer that becomes the next layer's target
// ---------------------------------------------------------------------------
__global__ __launch_bounds__(256) void lg_accum_zero(float* __restrict__ acc,
                                                     const float* __restrict__ nxt,
                                                     float* __restrict__ zbuf) {
  int i = blockIdx.x * 256 + threadIdx.x;
  acc[i] += nxt[i];
  zbuf[i] = 0.0f;
}

// ---------------------------------------------------------------------------
// d_out[1..] = acc * 0.25  (user_emb rows then item_emb rows, return order)
// ---------------------------------------------------------------------------
__global__ __launch_bounds__(256) void lg_scale_out(const float* __restrict__ acc,
                                                    float* __restrict__ outp) {
  int i = blockIdx.x * 256 + threadIdx.x;
  outp[1 + i] = acc[i] * 0.25f;
}

// ---------------------------------------------------------------------------
// Fused GEMM (f32 WMMA 16x16x4) + softplus loss.
// Block computes a 128x128 tile of scores = (acc_u/4) @ (acc_i/4)^T.
// 8 waves; wave w owns rows [16w,16w+16), 8 N-tiles of 16.
// LDS: A/B tiles 128x64 f32 each = 64 KB total, XOR-swizzled on k so the
// WMMA fragment reads (b64 per lane) are bank-conflict free.
// WMMA f32 16x16x4 operand layout (ISA 7.12.2):
//   A (16x4): lanes 0-15 hold row M=lane, {K0,K1}; lanes 16-31 hold {K2,K3}.
//   B (4x16): lanes 0-15 hold col N=lane, {K0,K1}; lanes 16-31 hold {K2,K3}.
//   (B[k][n] = item[n][k], so both fragments gather identically row-major.)
//   C/D: VGPR r -> M = r (+8 for lanes 16-31), N = lane & 15.
// ---------------------------------------------------------------------------
__global__ __launch_bounds__(256) void lg_gemm_loss(const float* __restrict__ acc,
                                                    const float* __restrict__ labels,
                                                    float* __restrict__ loss) {
  __shared__ float As[128 * 64];
  __shared__ float Bs[128 * 64];

  const float* __restrict__ A = acc;                        // user rows
  const float* __restrict__ B = acc + (size_t)UDIM * DDIM;  // item rows
  int row0 = blockIdx.y * 128;
  int col0 = blockIdx.x * 128;

  // Cooperative load: 128 rows x 16 float4 each, swizzle phys_k = k ^ ((r&15)<<2)
  for (int idx = threadIdx.x; idx < 128 * 16; idx += 256) {
    int r  = idx >> 4;
    int c4 = (idx & 15) << 2;
    int sw = c4 ^ ((r & 15) << 2);
    *(float4*)&As[r * 64 + sw] = *(const float4*)(A + (size_t)(row0 + r) * 64 + c4);
    *(float4*)&Bs[r * 64 + sw] = *(const float4*)(B + (size_t)(col0 + r) * 64 + c4);
  }
  __syncthreads();

  int wave  = threadIdx.x >> 5;
  int lane  = threadIdx.x & 31;
  int lrow  = lane & 15;
  int khalf = (lane >> 4) << 1;          // 0 for lanes 0-15, 2 for lanes 16-31
  int ra    = wave * 16 + lrow;          // A fragment row within block tile
  int swz   = lrow << 2;                 // same swizzle term for A and B reads

  v8f c[8] = {};                         // 8 N-tiles of 16x16 f32 accumulators

#pragma unroll
  for (int k0 = 0; k0 < 64; k0 += 4) {
    int ksw = (k0 + khalf) ^ swz;
    v2f a = *(const v2f*)&As[ra * 64 + ksw];
#pragma unroll
    for (int j = 0; j < 8; ++j) {
      v2f b = *(const v2f*)&Bs[(j * 16 + lrow) * 64 + ksw];
      c[j] = __builtin_amdgcn_wmma_f32_16x16x4_f32(
          /*neg_a=*/false, a, /*neg_b=*/false, b,
          /*c_mod=*/(short)0, c[j], /*reuse_a=*/false, /*reuse_b=*/false);
    }
  }

  // Fused loss: softplus(s) - s*label, s = dot * (1/16) (both factors carry /4)
  float lsum = 0.0f;
  int mbase = row0 + wave * 16 + ((lane >> 4) << 3);
#pragma unroll
  for (int j = 0; j < 8; ++j) {
    int n = col0 + j * 16 + lrow;
#pragma unroll
    for (int vr = 0; vr < 8; ++vr) {
      float s   = c[j][vr] * 0.0625f;
      float lab = labels[(size_t)(mbase + vr) * IDIM + n];
      float sp  = fmaxf(s, 0.0f) + log1pf(expf(-fabsf(s)));
      lsum += sp - s * lab;
    }
  }

  // wave32 reduction, one atomic per wave
#pragma unroll
  for (int off = 16; off > 0; off >>= 1)
    lsum += __shfl_down(lsum, off, 32);
  if (lane == 0) atomicAdd(loss, lsum);
}

__global__ void lg_finalize(const float* __restrict__ loss, float* __restrict__ outp) {
  outp[0] = loss[0] * (1.0f / ((float)UDIM * (float)IDIM));
}

// ---------------------------------------------------------------------------
extern "C" void kernel_launch(void* const* d_in, const int* in_sizes, int n_in,
                              void* d_out, int out_size, void* d_ws, size_t ws_size,
                              hipStream_t stream) {
  const float* users  = (const float*)d_in[0];
  const float* items  = (const float*)d_in[1];
  const int*   erow   = (const int*)d_in[2];
  const int*   ecol   = (const int*)d_in[3];
  const float* eval   = (const float*)d_in[4];
  const float* labels = (const float*)d_in[5];
  int E = in_sizes[2];

  float* ws   = (float*)d_ws;
  float* cur  = ws;
  float* nxt  = ws + (size_t)NELEM;
  float* acc  = ws + 2 * (size_t)NELEM;
  float* loss = ws + 3 * (size_t)NELEM;
  float* outp = (float*)d_out;

  lg_init<<<NELEM / 256, 256, 0, stream>>>(users, items, cur, acc, nxt, loss);

  for (int layer = 0; layer < 3; ++layer) {
    int spmm_blocks = (E * 32 + 255) / 256;  // one wave32 per edge
    lg_spmm<<<spmm_blocks, 256, 0, stream>>>(erow, ecol, eval, cur, nxt, E);
    lg_accum_zero<<<NELEM / 256, 256, 0, stream>>>(acc, nxt, cur);
    float* t = cur; cur = nxt; nxt = t;
  }

  lg_scale_out<<<NELEM / 256, 256, 0, stream>>>(acc, outp);
  lg_gemm_loss<<<dim3(64, 64), 256, 0, stream>>>(acc, labels, loss);
  lg_finalize<<<1, 1, 0, stream>>>(loss, outp);
}